// FRCLoss_71416716197913
// MI455X (gfx1250) — compile-verified
//
#include <hip/hip_runtime.h>
#include <math.h>

#define IMG   256
#define NPIX  (IMG * IMG)
#define CH    8                 // image pairs per chunk (192 / 8 = 24 chunks)

typedef __attribute__((ext_vector_type(2))) float v2f;
typedef __attribute__((ext_vector_type(8))) float v8f;

// D = A(16x4 f32) * B(4x16 f32) + C(16x16 f32), wave32 WMMA
__device__ __forceinline__ v8f wmma_f32(v2f a, v2f b, v8f c) {
    return __builtin_amdgcn_wmma_f32_16x16x4_f32(
        /*neg_a=*/false, a, /*neg_b=*/false, b,
        /*c_mod=*/(short)0, c, /*reuse_a=*/false, /*reuse_b=*/false);
}

// ---------------------------------------------------------------- twiddles
// W[j][k] = exp(-2*pi*i * (j*k mod 256) / 256)
__global__ void __launch_bounds__(256)
twiddle_kernel(float* __restrict__ Wr, float* __restrict__ Wi) {
    int idx = blockIdx.x * blockDim.x + threadIdx.x;   // 0..65535
    int j = idx >> 8, k = idx & 255;
    int t = (j * k) & 255;
    float ang = -6.283185307179586f * (float)t / 256.0f;
    float s, c;
    sincosf(ang, &s, &c);
    Wr[idx] = c;
    Wi[idx] = s;
}

__global__ void zero_kernel(float* __restrict__ p, int n) {
    int i = blockIdx.x * blockDim.x + threadIdx.x;
    if (i < n) p[i] = 0.f;
}

// ---------------------------------------------------------------- T = W * X
// X real (CH images), T complex. grid = (2, 16, CH), block = 256 (8 waves).
// Each wave computes one 16x16 tile of T_re and T_im.
__global__ void __launch_bounds__(256)
dft_rows_kernel(const float* __restrict__ X,
                const float* __restrict__ Wr, const float* __restrict__ Wi,
                float* __restrict__ Tr, float* __restrict__ Ti) {
    const int lane = threadIdx.x & 31;
    const int wave = threadIdx.x >> 5;
    const int half = lane >> 4;          // 0: lanes 0-15, 1: lanes 16-31
    const int l16  = lane & 15;
    const int n0   = (blockIdx.x * 8 + wave) * 16;
    const int m0   = blockIdx.y * 16;
    const int img  = blockIdx.z;

    const float* x  = X + (size_t)img * NPIX;
    const int arow  = (m0 + l16) * IMG;   // A-matrix row for this lane (W)
    const int bcol  = n0 + l16;           // B-matrix column (X)

    v8f accR = {}; v8f accI = {};
    for (int k0 = 0; k0 < IMG; k0 += 4) {
        const int ka = k0 + 2 * half;
        v2f aR = { Wr[arow + ka], Wr[arow + ka + 1] };
        v2f aI = { Wi[arow + ka], Wi[arow + ka + 1] };
        v2f b  = { x[ka * IMG + bcol], x[(ka + 1) * IMG + bcol] };
        accR = wmma_f32(aR, b, accR);
        accI = wmma_f32(aI, b, accI);
    }
    float* tr = Tr + (size_t)img * NPIX;
    float* ti = Ti + (size_t)img * NPIX;
#pragma unroll
    for (int j = 0; j < 8; ++j) {
        const int row = m0 + j + 8 * half;
        tr[row * IMG + n0 + l16] = accR[j];
        ti[row * IMG + n0 + l16] = accI[j];
    }
}

// ---------------------------------------------------------------- F = T * W
// T complex, W complex. Fre = Tr*Wr - Ti*Wi ; Fim = Tr*Wi + Ti*Wr.
__global__ void __launch_bounds__(256)
dft_cols_kernel(const float* __restrict__ Tr, const float* __restrict__ Ti,
                const float* __restrict__ Wr, const float* __restrict__ Wi,
                float* __restrict__ Fr, float* __restrict__ Fi) {
    const int lane = threadIdx.x & 31;
    const int wave = threadIdx.x >> 5;
    const int half = lane >> 4;
    const int l16  = lane & 15;
    const int n0   = (blockIdx.x * 8 + wave) * 16;
    const int m0   = blockIdx.y * 16;
    const int img  = blockIdx.z;

    const float* tr = Tr + (size_t)img * NPIX;
    const float* ti = Ti + (size_t)img * NPIX;
    const int arow  = (m0 + l16) * IMG;
    const int bcol  = n0 + l16;

    v8f a_rr = {}, a_ii = {}, a_ri = {}, a_ir = {};
    for (int k0 = 0; k0 < IMG; k0 += 4) {
        const int ka = k0 + 2 * half;
        v2f aR = { tr[arow + ka], tr[arow + ka + 1] };
        v2f aI = { ti[arow + ka], ti[arow + ka + 1] };
        v2f bR = { Wr[ka * IMG + bcol], Wr[(ka + 1) * IMG + bcol] };
        v2f bI = { Wi[ka * IMG + bcol], Wi[(ka + 1) * IMG + bcol] };
        a_rr = wmma_f32(aR, bR, a_rr);
        a_ii = wmma_f32(aI, bI, a_ii);   // f32 WMMA NEG is C-neg only: subtract at store
        a_ri = wmma_f32(aR, bI, a_ri);
        a_ir = wmma_f32(aI, bR, a_ir);
    }
    float* fr = Fr + (size_t)img * NPIX;
    float* fi = Fi + (size_t)img * NPIX;
#pragma unroll
    for (int j = 0; j < 8; ++j) {
        const int row = m0 + j + 8 * half;
        fr[row * IMG + n0 + l16] = a_rr[j] - a_ii[j];
        fi[row * IMG + n0 + l16] = a_ri[j] + a_ir[j];
    }
}

// ------------------------------------------------- per-segment accumulation
// One block per radius segment; seg_ids are sorted ascending, so binary-search
// the range. Block-tree reduction + single-writer "+=" => fully deterministic.
__device__ __forceinline__ int lower_bound_seg(const int* __restrict__ seg,
                                               int n, int v) {
    int lo = 0, hi = n;
    while (lo < hi) {
        int mid = (lo + hi) >> 1;
        if (seg[mid] < v) lo = mid + 1; else hi = mid;
    }
    return lo;
}

__global__ void __launch_bounds__(256)
gather_kernel(const float* __restrict__ F1r, const float* __restrict__ F1i,
              const float* __restrict__ F2r, const float* __restrict__ F2i,
              const int* __restrict__ rr, const int* __restrict__ cc,
              const int* __restrict__ seg, int npts, int nimgs,
              float* __restrict__ segsums) {
    const int s = blockIdx.x;                    // segment id
    const int t = threadIdx.x;
    const int start = lower_bound_seg(seg, npts, s);
    const int end   = lower_bound_seg(seg, npts, s + 1);

    float cR = 0.f, cI = 0.f, p1 = 0.f, p2 = 0.f;
    for (int p = start + t; p < end; p += blockDim.x) {
        const int off = rr[p] * IMG + cc[p];
        for (int i = 0; i < nimgs; ++i) {
            const int o = i * NPIX + off;
            const float ar = F1r[o], ai = F1i[o];
            const float br = F2r[o], bi = F2i[o];
            cR += ar * br + ai * bi;     // Re(p1 * conj(p2))
            cI += ai * br - ar * bi;     // Im(p1 * conj(p2))
            p1 += ar * ar + ai * ai;
            p2 += br * br + bi * bi;
        }
    }
    __shared__ float red[4][256];
    red[0][t] = cR; red[1][t] = cI; red[2][t] = p1; red[3][t] = p2;
    __syncthreads();
    for (int st = 128; st > 0; st >>= 1) {
        if (t < st) {
#pragma unroll
            for (int j = 0; j < 4; ++j) red[j][t] += red[j][t + st];
        }
        __syncthreads();
    }
    if (t == 0) {
#pragma unroll
        for (int j = 0; j < 4; ++j) segsums[s * 4 + j] += red[j][0];
    }
}

// ---------------------------------------------------------------- finalize
__global__ void __launch_bounds__(128)
finalize_kernel(const float* __restrict__ segsums, const float* __restrict__ w,
                float* __restrict__ out, int R, float scale) {
    __shared__ float red[128];
    const int t = threadIdx.x;
    float v = 0.f;
    if (t < R) {
        float curve;
        if (t == 0) {
            curve = 1.f;
        } else {
            const float crr = segsums[(t - 1) * 4 + 0];
            const float cii = segsums[(t - 1) * 4 + 1];
            const float pw1 = segsums[(t - 1) * 4 + 2];
            const float pw2 = segsums[(t - 1) * 4 + 3];
            const float num = sqrtf(crr * crr + cii * cii);
            const float den = fmaxf(pw1 * pw2, 1e-38f);
            curve = num * rsqrtf(den);
        }
        v = curve * w[t];
    }
    red[t] = v;
    __syncthreads();
    for (int st = 64; st > 0; st >>= 1) {
        if (t < st) red[t] += red[t + st];
        __syncthreads();
    }
    if (t == 0) out[0] = scale * red[0];
}

// ---------------------------------------------------------------- launcher
extern "C" void kernel_launch(void* const* d_in, const int* in_sizes, int n_in,
                              void* d_out, int out_size, void* d_ws, size_t ws_size,
                              hipStream_t stream) {
    const float* input   = (const float*)d_in[0];   // (64,3,256,256) f32
    const float* target  = (const float*)d_in[1];   // (64,3,256,256) f32
    const float* weights = (const float*)d_in[2];   // (127,) f32
    const int*   rr      = (const int*)d_in[3];
    const int*   cc      = (const int*)d_in[4];
    const int*   seg     = (const int*)d_in[5];

    const int R     = in_sizes[2];          // 127
    const int nseg  = R - 1;                // 126
    const int npts  = in_sizes[3];
    const int nimg  = in_sizes[0] / NPIX;   // 192

    // workspace layout (floats): ~13.1 MB total
    float* ws   = (float*)d_ws;
    float* Wr   = ws;
    float* Wi   = Wr  + NPIX;
    float* Tre  = Wi  + NPIX;
    float* Tim  = Tre + (size_t)CH * NPIX;
    float* F1r  = Tim + (size_t)CH * NPIX;
    float* F1i  = F1r + (size_t)CH * NPIX;
    float* F2r  = F1i + (size_t)CH * NPIX;
    float* F2i  = F2r + (size_t)CH * NPIX;
    float* segs = F2i + (size_t)CH * NPIX;  // nseg*4 floats

    twiddle_kernel<<<NPIX / 256, 256, 0, stream>>>(Wr, Wi);
    zero_kernel<<<(nseg * 4 + 255) / 256, 256, 0, stream>>>(segs, nseg * 4);

    const dim3 mmGrid(2, 16, CH);     // 2*8 waves -> 16 n-tiles, 16 m-tiles, CH imgs
    const int  nChunks = nimg / CH;
    for (int c = 0; c < nChunks; ++c) {
        const size_t base = (size_t)c * CH * NPIX;
        dft_rows_kernel<<<mmGrid, 256, 0, stream>>>(input + base, Wr, Wi, Tre, Tim);
        dft_cols_kernel<<<mmGrid, 256, 0, stream>>>(Tre, Tim, Wr, Wi, F1r, F1i);
        dft_rows_kernel<<<mmGrid, 256, 0, stream>>>(target + base, Wr, Wi, Tre, Tim);
        dft_cols_kernel<<<mmGrid, 256, 0, stream>>>(Tre, Tim, Wr, Wi, F2r, F2i);
        gather_kernel<<<nseg, 256, 0, stream>>>(F1r, F1i, F2r, F2i,
                                                rr, cc, seg, npts, CH, segs);
    }
    finalize_kernel<<<1, 128, 0, stream>>>(segs, weights, (float*)d_out,
                                           R, (float)nimg);
}